// GAT_13134009991658
// MI455X (gfx1250) — compile-verified
//
#include <hip/hip_runtime.h>
#include <hip/hip_bf16.h>

typedef __attribute__((ext_vector_type(2))) float v2f;
typedef __attribute__((ext_vector_type(8))) float v8f;

#define DHID 128          // feature width (Din == H == 128)
#define NEG_SLOPE 0.2f
#define ORD_NEG_INF 0x007FFFFFu   // f2ord(-inf)

// ---- monotonic float<->uint mapping for atomic max on signed floats ----
__device__ __forceinline__ unsigned f2ord(float f) {
  unsigned u = __float_as_uint(f);
  return (u & 0x80000000u) ? ~u : (u | 0x80000000u);
}
__device__ __forceinline__ float ord2f(unsigned u) {
  return (u & 0x80000000u) ? __uint_as_float(u & 0x7FFFFFFFu)
                           : __uint_as_float(~u);
}

// ============================================================
// GEMM + bias: C[N,128] = A[N,128] @ W[128,128] + bias
// one wave -> 16 rows x 128 cols via 8 f32 WMMA accumulators
// ============================================================
__global__ __launch_bounds__(256) void gemm_bias_wmma(
    const float* __restrict__ A, const float* __restrict__ W,
    const float* __restrict__ bias, float* __restrict__ C, int nRowTiles) {
  const int lane = threadIdx.x & 31;
  const int wave = threadIdx.x >> 5;
  const int rowTile = blockIdx.x * 8 + wave;
  if (rowTile >= nRowTiles) return;        // whole-wave exit: EXEC stays all-1s
  const int m0 = rowTile * 16;
  const int lm = lane & 15;                // M (A) / N (B) position within tile
  const int kq = (lane >> 4) * 2;          // K sub-offset: 0 (lanes 0-15) or 2

  v8f acc[8];
  const v8f vzero = {0.f, 0.f, 0.f, 0.f, 0.f, 0.f, 0.f, 0.f};
#pragma unroll
  for (int j = 0; j < 8; ++j) acc[j] = vzero;

  const float* arow = A + (size_t)(m0 + lm) * DHID;

  for (int kk = 0; kk < DHID; kk += 4) {
    // prefetch the A stream ahead (gfx1250 global_prefetch_b8)
    if ((kk & 31) == 0) __builtin_prefetch(arow + kk + 64, 0, 0);

    // A fragment: 16x4 f32 — lanes 0-15 hold K=kk,kk+1; lanes 16-31 K=kk+2,kk+3
    v2f a;
    a.x = arow[kk + kq];
    a.y = arow[kk + kq + 1];

    const float* w0 = W + (size_t)(kk + kq) * DHID + lm;   // row K=kk+kq
    const float* w1 = w0 + DHID;                           // row K=kk+kq+1
#pragma unroll
    for (int j = 0; j < 8; ++j) {
      // B fragment: 4x16 f32, N = j*16 + lm
      v2f b;
      b.x = w0[j * 16];
      b.y = w1[j * 16];
      acc[j] = __builtin_amdgcn_wmma_f32_16x16x4_f32(
          false, a, false, b, (short)0, acc[j], false, false);
    }
  }

  // C/D layout: VGPR r -> lanes 0-15: M=m0+r ; lanes 16-31: M=m0+8+r ; N=lm
  const int rbase = m0 + ((lane >> 4) << 3);
#pragma unroll
  for (int j = 0; j < 8; ++j) {
    const int col = j * 16 + lm;
    const float bv = bias[col];
#pragma unroll
    for (int r = 0; r < 8; ++r)
      C[(size_t)(rbase + r) * DHID + col] = acc[j][r] + bv;
  }
}

// ============================================================
// init: zero the output accumulator, reset segment max / denom
// ============================================================
__global__ __launch_bounds__(256) void init_state(
    float* __restrict__ accOut, unsigned* __restrict__ m,
    float* __restrict__ denom, int nh, int n) {
  const int i = blockIdx.x * 256 + threadIdx.x;
  if (i < nh) accOut[i] = 0.f;
  if (i < n) { m[i] = ORD_NEG_INF; denom[i] = 0.f; }
}

// ============================================================
// per-edge attention logit: one wave per edge
// logit = sum_h leaky_relu(xl[src][h] + xr[dst][h]) * att[h]
// ============================================================
__global__ __launch_bounds__(256) void edge_logits(
    const float* __restrict__ xl, const float* __restrict__ xr,
    const float* __restrict__ att, const long long* __restrict__ eidx,
    float* __restrict__ logits, unsigned* __restrict__ m, int E) {
  const int lane = threadIdx.x & 31;
  const int e = blockIdx.x * 8 + (threadIdx.x >> 5);
  if (e >= E) return;
  const int s = (int)eidx[e];
  const int d = (int)eidx[(size_t)E + e];

  const float4 l = *(const float4*)(xl + (size_t)s * DHID + lane * 4);
  const float4 r = *(const float4*)(xr + (size_t)d * DHID + lane * 4);
  const float4 a = *(const float4*)(att + lane * 4);

  float acc = 0.f, v;
  v = l.x + r.x; acc += (v > 0.f ? v : NEG_SLOPE * v) * a.x;
  v = l.y + r.y; acc += (v > 0.f ? v : NEG_SLOPE * v) * a.y;
  v = l.z + r.z; acc += (v > 0.f ? v : NEG_SLOPE * v) * a.z;
  v = l.w + r.w; acc += (v > 0.f ? v : NEG_SLOPE * v) * a.w;

#pragma unroll
  for (int off = 16; off > 0; off >>= 1) acc += __shfl_xor(acc, off, 32);

  if (lane == 0) {
    logits[e] = acc;
    atomicMax(&m[d], f2ord(acc));
  }
}

// ============================================================
// p = exp(logit - m[dst]);  denom[dst] += p   (in-place over logits)
// ============================================================
__global__ __launch_bounds__(256) void edge_exp(
    float* __restrict__ logits, const unsigned* __restrict__ m,
    float* __restrict__ denom, const long long* __restrict__ eidx, int E) {
  const int e = blockIdx.x * 256 + threadIdx.x;
  if (e >= E) return;
  const int d = (int)eidx[(size_t)E + e];
  const float p = __expf(logits[e] - ord2f(m[d]));
  logits[e] = p;
  unsafeAtomicAdd(&denom[d], p);
}

// ============================================================
// out[dst] += (p/denom[dst]) * xl[src] : one wave per edge, fp32 HW atomics
// ============================================================
__global__ __launch_bounds__(256) void edge_aggr(
    const float* __restrict__ p, const float* __restrict__ denom,
    const float* __restrict__ xl, const long long* __restrict__ eidx,
    float* __restrict__ out, int E) {
  const int lane = threadIdx.x & 31;
  const int e = blockIdx.x * 8 + (threadIdx.x >> 5);
  if (e >= E) return;
  const int s = (int)eidx[e];
  const int d = (int)eidx[(size_t)E + e];
  const float alpha = p[e] / fmaxf(denom[d], 1e-16f);

  const float4 v = *(const float4*)(xl + (size_t)s * DHID + lane * 4);
  float* o = out + (size_t)d * DHID + lane * 4;
  unsafeAtomicAdd(o + 0, alpha * v.x);
  unsafeAtomicAdd(o + 1, alpha * v.y);
  unsafeAtomicAdd(o + 2, alpha * v.z);
  unsafeAtomicAdd(o + 3, alpha * v.w);
}

// ============================================================
// buf = relu(buf + bias)  (in place)
// ============================================================
__global__ __launch_bounds__(256) void finalize_relu(
    float* __restrict__ buf, const float* __restrict__ bias, int nh) {
  const int i = blockIdx.x * 256 + threadIdx.x;
  if (i >= nh) return;
  buf[i] = fmaxf(buf[i] + bias[i & (DHID - 1)], 0.f);
}

extern "C" void kernel_launch(void* const* d_in, const int* in_sizes, int n_in,
                              void* d_out, int out_size, void* d_ws, size_t ws_size,
                              hipStream_t stream) {
  const float* x        = (const float*)d_in[0];
  const long long* eidx = (const long long*)d_in[1];
  const float* Wl1  = (const float*)d_in[2];
  const float* bl1  = (const float*)d_in[3];
  const float* Wr1  = (const float*)d_in[4];
  const float* br1  = (const float*)d_in[5];
  const float* att1 = (const float*)d_in[6];
  const float* b1   = (const float*)d_in[7];
  const float* Wl2  = (const float*)d_in[8];
  const float* bl2  = (const float*)d_in[9];
  const float* Wr2  = (const float*)d_in[10];
  const float* br2  = (const float*)d_in[11];
  const float* att2 = (const float*)d_in[12];
  const float* b2   = (const float*)d_in[13];

  const int N = in_sizes[0] / DHID;
  const int E = in_sizes[1] / 2;
  const size_t NH = (size_t)N * DHID;

  float* ws    = (float*)d_ws;
  float* xl    = ws;
  float* xr    = xl + NH;
  float* hbuf  = xr + NH;
  float* lg    = hbuf + NH;
  unsigned* mm = (unsigned*)(lg + E);
  float* denom = (float*)(mm + N);
  float* out   = (float*)d_out;

  const int rowTiles   = N / 16;
  const int gemmBlks   = (rowTiles + 7) / 8;
  const int edgeBlks8  = (E + 7) / 8;
  const int edgeBlks1  = (E + 255) / 256;
  const int nhBlks     = ((int)NH + 255) / 256;
  const dim3 blk(256);

  // ---------------- layer 1: accumulate into hbuf ----------------
  init_state<<<nhBlks, blk, 0, stream>>>(hbuf, mm, denom, (int)NH, N);
  gemm_bias_wmma<<<gemmBlks, blk, 0, stream>>>(x, Wl1, bl1, xl, rowTiles);
  gemm_bias_wmma<<<gemmBlks, blk, 0, stream>>>(x, Wr1, br1, xr, rowTiles);
  edge_logits<<<edgeBlks8, blk, 0, stream>>>(xl, xr, att1, eidx, lg, mm, E);
  edge_exp<<<edgeBlks1, blk, 0, stream>>>(lg, mm, denom, eidx, E);
  edge_aggr<<<edgeBlks8, blk, 0, stream>>>(lg, denom, xl, eidx, hbuf, E);
  finalize_relu<<<nhBlks, blk, 0, stream>>>(hbuf, b1, (int)NH);

  // ---------------- layer 2: accumulate into d_out ----------------
  init_state<<<nhBlks, blk, 0, stream>>>(out, mm, denom, (int)NH, N);
  gemm_bias_wmma<<<gemmBlks, blk, 0, stream>>>(hbuf, Wl2, bl2, xl, rowTiles);
  gemm_bias_wmma<<<gemmBlks, blk, 0, stream>>>(hbuf, Wr2, br2, xr, rowTiles);
  edge_logits<<<edgeBlks8, blk, 0, stream>>>(xl, xr, att2, eidx, lg, mm, E);
  edge_exp<<<edgeBlks1, blk, 0, stream>>>(lg, mm, denom, eidx, E);
  edge_aggr<<<edgeBlks8, blk, 0, stream>>>(lg, denom, xl, eidx, out, E);
  finalize_relu<<<nhBlks, blk, 0, stream>>>(out, b2, (int)NH);
}